// DGCNN_partseg_16509854285877
// MI455X (gfx1250) — compile-verified
//
#include <hip/hip_runtime.h>
#include <hip/hip_bf16.h>

// ---------------------------------------------------------------------------
// DGCNN part-seg forward on gfx1250 (CDNA5).
// All conv/FC layers -> one generic WMMA f32 16x16x4 GEMM kernel (templated
// on activation / epilogue mode) with fused BN (*g + b), LeakyReLU(0.2),
// and optional grouped atomic-max epilogue. Each wave computes a 16(O)x64(M)
// tile; A operand reused across 4 WMMAs per K-step. Edge tiles use clamped
// loads + v_cndmask selects (no EXEC juggling).
// ---------------------------------------------------------------------------

typedef float v2f __attribute__((ext_vector_type(2)));
typedef float v8f __attribute__((ext_vector_type(8)));

#define NEG_INF (-3.402823466e+38f)

__device__ __forceinline__ void atomicMaxF(float* addr, float val) {
    unsigned int* ua = (unsigned int*)addr;
    unsigned int old = *((volatile unsigned int*)ua);
    while (__uint_as_float(old) < val) {
        unsigned int assumed = old;
        old = atomicCAS(ua, assumed, __float_as_uint(val));
        if (old == assumed) break;
    }
}

// ---------------------------------------------------------------------------
// WMMA GEMM: Y[b,o,m] = act( (sum_c W[o,c] * X[b,c,m]) * g[o] + bias[o] )
//  ACT: 0=none, 1=lrelu(0.2)
//  GMODE: 0=direct store (B,O,M); 1=atomic-max over groups of groupReduce
//         -> (B,O,M/groupReduce); 2=same but transposed out[(o*G+grp)*B+b]
// f32 WMMA 16x16x4 per ISA VGPR layout:
//   A 16x4: lanes 0-15 hold K=0,1 ; lanes 16-31 hold K=2,3 (rows M=lane&15)
//   B 4x16: same K split across lane halves, N = lane&15
//   C/D   : vgpr r -> rows r (lanes 0-15) and r+8 (lanes 16-31)
// ---------------------------------------------------------------------------
template <int ACT, int GMODE>
__global__ void wmma_gemm_bn_act(const float* __restrict__ W,
                                 const float* __restrict__ X,
                                 const float* __restrict__ g,
                                 const float* __restrict__ bias,
                                 float* __restrict__ Y,
                                 int O, int C, int M,
                                 long long xStride, long long yStride,
                                 int groupReduce) {
    int mT = blockIdx.x * 64;
    int oT = blockIdx.y * 16;
    int b  = blockIdx.z;
    const float* Xb = X + (size_t)b * (size_t)xStride;
    int lane = threadIdx.x & 31;
    int half = lane >> 4;
    int l    = lane & 15;
    int row  = oT + l;   // W row (output channel) this lane loads
    int col  = mT + l;   // first X column (spatial) this lane loads
    v8f acc[4];
    acc[0] = (v8f){0.f,0.f,0.f,0.f,0.f,0.f,0.f,0.f};
    acc[1] = acc[0]; acc[2] = acc[0]; acc[3] = acc[0];

    const bool fast = (mT + 64 <= M) && (oT + 16 <= O) && ((C & 3) == 0);
    if (fast) {
        const float* wrow = W + (size_t)row * C;
        for (int c0 = 0; c0 < C; c0 += 4) {
            int ca = c0 + half * 2;
            v2f a = *(const v2f*)(wrow + ca);          // 8B aligned: C%4==0, ca even
            const float* xr0 = Xb + (size_t)ca * M + col;
            const float* xr1 = Xb + (size_t)(ca + 1) * M + col;
#pragma unroll
            for (int s = 0; s < 4; ++s) {
                v2f bb;
                bb.x = xr0[16 * s];
                bb.y = xr1[16 * s];
                acc[s] = __builtin_amdgcn_wmma_f32_16x16x4_f32(false, a, false, bb,
                                                               (short)0, acc[s], false, false);
            }
        }
    } else {
        // Edge tiles: clamp indices so loads are always in-bounds, then mask
        // with selects (v_cndmask) instead of EXEC-predicated loads.
        int rowc = min(row, O - 1);
        bool rv = (row < O);
        for (int c0 = 0; c0 < C; c0 += 4) {
            int ca   = c0 + half * 2;
            int ca0  = min(ca, C - 1);
            int ca1  = min(ca + 1, C - 1);
            bool c0v = (ca < C);
            bool c1v = (ca + 1 < C);
            float w0 = W[(size_t)rowc * C + ca0];
            float w1 = W[(size_t)rowc * C + ca1];
            v2f a;
            a.x = (rv && c0v) ? w0 : 0.f;
            a.y = (rv && c1v) ? w1 : 0.f;
            const float* xr0 = Xb + (size_t)ca0 * M;
            const float* xr1 = Xb + (size_t)ca1 * M;
#pragma unroll
            for (int s = 0; s < 4; ++s) {
                int cc  = col + 16 * s;
                int ccc = min(cc, M - 1);
                float b0 = xr0[ccc];
                float b1 = xr1[ccc];
                bool mv = (cc < M);
                v2f bb;
                bb.x = (mv && c0v) ? b0 : 0.f;
                bb.y = (mv && c1v) ? b1 : 0.f;
                acc[s] = __builtin_amdgcn_wmma_f32_16x16x4_f32(false, a, false, bb,
                                                               (short)0, acc[s], false, false);
            }
        }
    }

    int G = (GMODE > 0) ? (M / groupReduce) : 0;
#pragma unroll
    for (int s = 0; s < 4; ++s) {
        int m = col + 16 * s;
        if (m >= M) continue;
#pragma unroll
        for (int r = 0; r < 8; ++r) {
            int o = oT + r + half * 8;
            if (o < O) {
                float y = acc[s][r];
                if (g)    y *= g[o];
                if (bias) y += bias[o];
                if (ACT == 1) y = (y > 0.f) ? y : 0.2f * y;
                if (GMODE == 0) {
                    Y[(size_t)b * yStride + (size_t)o * M + m] = y;
                } else {
                    int grp = m / groupReduce;
                    float* dst = (GMODE == 2)
                        ? (Y + (size_t)(o * G + grp) * gridDim.z + b)
                        : (Y + (size_t)b * yStride + (size_t)o * G + grp);
                    atomicMaxF(dst, y);
                }
            }
        }
    }
}

// ---------------------------------------------------------------------------
// KNN: for each (b, nq) find top-k of d[m] = 2*<q,p_m> - |q|^2 - |p_m|^2.
// One 256-thread block per query point; distances in LDS; k rounds of
// tie-stable (lowest index first, matching lax.top_k) argmax reduction.
// ---------------------------------------------------------------------------
__global__ void knn_kernel(const float* __restrict__ q, const float* __restrict__ p,
                           int C, int Nq, int Np, int k, int* __restrict__ idx) {
    __shared__ float qs[64];
    __shared__ float d[2048];
    __shared__ float sv[256];
    __shared__ int   si[256];
    int b  = blockIdx.x / Nq;
    int nq = blockIdx.x % Nq;
    int tid = threadIdx.x;
    const float* qb = q + (size_t)b * C * Nq;
    const float* pb = p + (size_t)b * C * Np;
    for (int c = tid; c < C; c += blockDim.x) qs[c] = qb[(size_t)c * Nq + nq];
    __syncthreads();
    float qq = 0.f;
    for (int c = 0; c < C; ++c) qq += qs[c] * qs[c];
    for (int m = tid; m < Np; m += blockDim.x) {
        float pp = 0.f, inner = 0.f;
        for (int c = 0; c < C; ++c) {
            float pv = pb[(size_t)c * Np + m];
            pp    += pv * pv;
            inner += qs[c] * pv;
        }
        d[m] = 2.f * inner - qq - pp;
    }
    __syncthreads();
    int* out = idx + ((size_t)b * Nq + nq) * k;
    for (int r = 0; r < k; ++r) {
        float bv = NEG_INF;
        int   bi = 0x7fffffff;
        for (int m = tid; m < Np; m += blockDim.x) {
            float v = d[m];
            if (v > bv || (v == bv && m < bi)) { bv = v; bi = m; }
        }
        sv[tid] = bv; si[tid] = bi;
        __syncthreads();
        for (int s = blockDim.x >> 1; s > 0; s >>= 1) {
            if (tid < s) {
                float ov = sv[tid + s]; int oi = si[tid + s];
                if (ov > sv[tid] || (ov == sv[tid] && oi < si[tid])) { sv[tid] = ov; si[tid] = oi; }
            }
            __syncthreads();
        }
        if (tid == 0) { out[r] = si[0]; d[si[0]] = NEG_INF; }
        __syncthreads();
    }
}

// graph_feature: out (B, 2C, N, K); c<C -> x[nbr]-x[ctr], else x[ctr]
__global__ void graph_feature_kernel(const float* __restrict__ x, const int* __restrict__ idx,
                                     float* __restrict__ out, int B, int C, int N, int K) {
    long long i = (long long)blockIdx.x * blockDim.x + threadIdx.x;
    long long tot = (long long)B * 2 * C * N * K;
    if (i >= tot) return;
    int kk = i % K;  long long t = i / K;
    int n  = t % N;  t /= N;
    int c  = t % (2 * C);
    int b  = (int)(t / (2 * C));
    float ctr = x[((long long)b * C + (c % C)) * N + n];
    float v;
    if (c < C) {
        int m = idx[((long long)b * N + n) * K + kk];
        v = x[((long long)b * C + c) * N + m] - ctr;
    } else {
        v = ctr;
    }
    out[i] = v;
}

// gather: out[b,c,j] = x[b,c,idx[b,j]]
__global__ void gather_kernel(const float* __restrict__ x, const int* __restrict__ idx,
                              float* __restrict__ out, int B, int C, int Nin, int Nout) {
    long long i = (long long)blockIdx.x * blockDim.x + threadIdx.x;
    long long tot = (long long)B * C * Nout;
    if (i >= tot) return;
    int j = i % Nout; long long t = i / Nout;
    int c = t % C;    int b = (int)(t / C);
    int src = idx[(long long)b * Nout + j];
    out[i] = x[((long long)b * C + c) * Nin + src];
}

// gather + max over k: out[b,c,j] = max_kk x[b,c,idx[b,j,kk]]
__global__ void gather_max_kernel(const float* __restrict__ x, const int* __restrict__ idx,
                                  float* __restrict__ out, int B, int C, int Nin, int Nout, int k) {
    long long i = (long long)blockIdx.x * blockDim.x + threadIdx.x;
    long long tot = (long long)B * C * Nout;
    if (i >= tot) return;
    int j = i % Nout; long long t = i / Nout;
    int c = t % C;    int b = (int)(t / C);
    const int* ip = idx + ((long long)b * Nout + j) * k;
    float best = NEG_INF;
    for (int kk = 0; kk < k; ++kk) {
        float v = x[((long long)b * C + c) * Nin + ip[kk]];
        best = fmaxf(best, v);
    }
    out[i] = best;
}

__global__ void fill_kernel(float* __restrict__ p, long long n, float v) {
    long long i = (long long)blockIdx.x * blockDim.x + threadIdx.x;
    if (i < n) p[i] = v;
}

// in-place a = relu(a + b)
__global__ void add_relu_kernel(float* __restrict__ a, const float* __restrict__ b, long long n) {
    long long i = (long long)blockIdx.x * blockDim.x + threadIdx.x;
    if (i < n) a[i] = fmaxf(a[i] + b[i], 0.f);
}

__global__ void max4_kernel(const float* a, const float* b, const float* c, const float* d,
                            float* __restrict__ o, long long n) {
    long long i = (long long)blockIdx.x * blockDim.x + threadIdx.x;
    if (i < n) o[i] = fmaxf(fmaxf(a[i], b[i]), fmaxf(c[i], d[i]));
}

// x' = einsum('bcn,bcd->bdn'); t stored as tbuf[(c*3+d)*B + b]
__global__ void transform_kernel(const float* __restrict__ x, const float* __restrict__ t,
                                 float* __restrict__ out, int B, int N) {
    long long i = (long long)blockIdx.x * blockDim.x + threadIdx.x;
    long long tot = (long long)B * 3 * N;
    if (i >= tot) return;
    int n = i % N; long long tt = i / N;
    int dd = tt % 3; int b = (int)(tt / 3);
    float s = 0.f;
    for (int c = 0; c < 3; ++c)
        s += x[((long long)b * 3 + c) * N + n] * t[(c * 3 + dd) * B + b];
    out[((long long)b * 3 + dd) * N + n] = s;
}

// out (B, C1+C2, N) = concat(a (B,C1,N), b (B,C2,N))
__global__ void cat2_kernel(const float* __restrict__ a, const float* __restrict__ bs,
                            float* __restrict__ out, int B, int C1, int C2, int N) {
    long long i = (long long)blockIdx.x * blockDim.x + threadIdx.x;
    long long tot = (long long)B * (C1 + C2) * N;
    if (i >= tot) return;
    int n = i % N; long long t = i / N;
    int c = t % (C1 + C2); int b = (int)(t / (C1 + C2));
    out[i] = (c < C1) ? a[((long long)b * C1 + c) * N + n]
                      : bs[((long long)b * C2 + (c - C1)) * N + n];
}

// out (B, 1152, N3): [0,1024)=gmax (bcast), [1024,1088)=lf (bcast), [1088,1152)=x3
__global__ void catg_kernel(const float* __restrict__ gmax, const float* __restrict__ lf,
                            const float* __restrict__ x3, float* __restrict__ out,
                            int B, int N3) {
    long long i = (long long)blockIdx.x * blockDim.x + threadIdx.x;
    long long tot = (long long)B * 1152 * N3;
    if (i >= tot) return;
    int n = i % N3; long long t = i / N3;
    int c = t % 1152; int b = (int)(t / 1152);
    float v;
    if (c < 1024)      v = gmax[(long long)b * 1024 + c];
    else if (c < 1088) v = lf[(long long)b * 64 + (c - 1024)];
    else               v = x3[((long long)b * 64 + (c - 1088)) * N3 + n];
    out[i] = v;
}

// ---------------------------------------------------------------------------
// Host orchestration
// ---------------------------------------------------------------------------
static inline unsigned gcd1(long long n) { return (unsigned)((n + 255) / 256); }

static void launch_gemm(const float* W, const float* X, const float* g, const float* b,
                        float* Y, int O, int C, int M, long long xS, long long yS,
                        int act, int grp, int tr, int batch, hipStream_t s) {
    dim3 grid((unsigned)((M + 63) / 64), (unsigned)((O + 15) / 16), (unsigned)batch);
    dim3 blk(32);
    if (grp > 0) {
        if (tr) wmma_gemm_bn_act<1, 2><<<grid, blk, 0, s>>>(W, X, g, b, Y, O, C, M, xS, yS, grp);
        else    wmma_gemm_bn_act<1, 1><<<grid, blk, 0, s>>>(W, X, g, b, Y, O, C, M, xS, yS, grp);
    } else if (act == 1) {
        wmma_gemm_bn_act<1, 0><<<grid, blk, 0, s>>>(W, X, g, b, Y, O, C, M, xS, yS, 0);
    } else {
        wmma_gemm_bn_act<0, 0><<<grid, blk, 0, s>>>(W, X, g, b, Y, O, C, M, xS, yS, 0);
    }
}

extern "C" void kernel_launch(void* const* d_in, const int* in_sizes, int n_in,
                              void* d_out, int out_size, void* d_ws, size_t ws_size,
                              hipStream_t stream) {
    (void)in_sizes; (void)n_in; (void)out_size; (void)ws_size;
    const int B = 8, N = 2048, KK = 40;
    const int N1 = 512, N2 = 128, N3 = 32;

    const float* x_in  = (const float*)d_in[0];
    const float* l_in  = (const float*)d_in[1];
    const int*   sidx1 = (const int*)d_in[2];
    const int*   sidx2 = (const int*)d_in[3];
    const int*   sidx3 = (const int*)d_in[4];
    int pi = 5;
    auto NXT = [&]() { return (const float*)d_in[pi++]; };
    // tnet (insertion order: c1,c2,c3,w4,g4,b4,w5,g5,b5,wt,bt)
    const float *c1w=NXT(),*c1g=NXT(),*c1b=NXT();
    const float *c2w=NXT(),*c2g=NXT(),*c2b=NXT();
    const float *c3w=NXT(),*c3g=NXT(),*c3b=NXT();
    const float *w4 =NXT(),*g4 =NXT(),*b4 =NXT();
    const float *w5 =NXT(),*g5 =NXT(),*b5 =NXT();
    const float *wt =NXT(),*bt =NXT();
    // ec0..ec3: two conv layers each
    const float *e0w0=NXT(),*e0g0=NXT(),*e0b0=NXT(),*e0w1=NXT(),*e0g1=NXT(),*e0b1=NXT();
    const float *e1w0=NXT(),*e1g0=NXT(),*e1b0=NXT(),*e1w1=NXT(),*e1g1=NXT(),*e1b1=NXT();
    const float *e2w0=NXT(),*e2g0=NXT(),*e2b0=NXT(),*e2w1=NXT(),*e2g1=NXT(),*e2b1=NXT();
    const float *e3w0=NXT(),*e3g0=NXT(),*e3b0=NXT(),*e3w1=NXT(),*e3g1=NXT(),*e3b1=NXT();
    // pn0..pn3 (single 64->1024 layer each)
    const float *p0w=NXT(),*p0g=NXT(),*p0b=NXT();
    const float *p1w=NXT(),*p1g=NXT(),*p1b=NXT();
    const float *p2w=NXT(),*p2g=NXT(),*p2b=NXT();
    const float *p3w=NXT(),*p3g=NXT(),*p3b=NXT();
    // label 16->64
    const float *lw=NXT(),*lg=NXT(),*lb=NXT();
    // pn4..pn7
    const float *p4w=NXT(),*p4g=NXT(),*p4b=NXT();
    const float *p5w=NXT(),*p5g=NXT(),*p5b=NXT();
    const float *p6w=NXT(),*p6g=NXT(),*p6b=NXT();
    const float *p7w=NXT(),*p7g=NXT(),*p7b=NXT();
    const float *w8 =NXT();

    // ---- workspace carve ----
    size_t off = 0;
    char* wsb = (char*)d_ws;
    auto ALLOC = [&](size_t bytes) { void* r = wsb + off; off += (bytes + 255) & ~(size_t)255; return r; };
    float* xT    = (float*)ALLOC((size_t)B*3*N*4);
    float* x0    = (float*)ALLOC((size_t)B*64*N*4);
    float* node1 = (float*)ALLOC((size_t)B*3*N1*4);
    float* f1    = (float*)ALLOC((size_t)B*64*N1*4);
    float* x1    = (float*)ALLOC((size_t)B*64*N1*4);
    float* node2 = (float*)ALLOC((size_t)B*3*N2*4);
    float* f2    = (float*)ALLOC((size_t)B*64*N2*4);
    float* x2    = (float*)ALLOC((size_t)B*64*N2*4);
    float* node3 = (float*)ALLOC((size_t)B*3*N3*4);
    float* f3    = (float*)ALLOC((size_t)B*64*N3*4);
    float* x3    = (float*)ALLOC((size_t)B*64*N3*4);
    float* xt0   = (float*)ALLOC((size_t)B*1024*4);
    float* xt1   = (float*)ALLOC((size_t)B*1024*4);
    float* xt2   = (float*)ALLOC((size_t)B*1024*4);
    float* xt3   = (float*)ALLOC((size_t)B*1024*4);
    float* gmax  = (float*)ALLOC((size_t)B*1024*4);
    float* lf    = (float*)ALLOC((size_t)B*64*4);
    float* h1024T= (float*)ALLOC((size_t)1024*B*4);
    float* fc4   = (float*)ALLOC((size_t)512*B*4);
    float* fc5   = (float*)ALLOC((size_t)256*B*4);
    float* tbuf  = (float*)ALLOC((size_t)9*B*4);
    float* h4    = (float*)ALLOC((size_t)B*256*N3*4);
    float* h5    = (float*)ALLOC((size_t)B*256*N2*4);
    float* h6    = (float*)ALLOC((size_t)B*256*N1*4);
    float* h7    = (float*)ALLOC((size_t)B*128*N*4);
    int*   idxb  = (int*)  ALLOC((size_t)B*N*KK*4);
    float* sA    = (float*)ALLOC((size_t)176 << 20);   // big stage scratch
    float* sB    = (float*)ALLOC((size_t)40  << 20);   // secondary stage scratch
    auto sAat = [&](size_t mb) { return sA + (mb << 20) / 4; };
    auto sBat = [&](size_t mb) { return sB + (mb << 20) / 4; };

    dim3 B256(256);
    auto FILL = [&](float* p, long long n) { fill_kernel<<<gcd1(n), B256, 0, stream>>>(p, n, NEG_INF); };

    // ===================== transform_net on original x =====================
    knn_kernel<<<B * N, B256, 0, stream>>>(x_in, x_in, 3, N, N, KK, idxb);
    float* xg  = sB;            // (B,6,N,K)  ~15.7MB
    float* c1o = sA;            // (B,64,N,K) ~168MB
    float* c2m = sB;            // (B,128,N)  ~8.4MB (xg dead by then)
    {
        long long tot = (long long)B * 6 * N * KK;
        graph_feature_kernel<<<gcd1(tot), B256, 0, stream>>>(x_in, idxb, xg, B, 3, N, KK);
    }
    launch_gemm(c1w, xg, c1g, c1b, c1o, 64, 6, N * KK,
                (long long)6 * N * KK, (long long)64 * N * KK, 1, 0, 0, B, stream);
    FILL(c2m, (long long)B * 128 * N);
    launch_gemm(c2w, c1o, c2g, c2b, c2m, 128, 64, N * KK,
                (long long)64 * N * KK, (long long)128 * N, 1, KK, 0, B, stream);
    FILL(h1024T, (long long)1024 * B);
    launch_gemm(c3w, c2m, c3g, c3b, h1024T, 1024, 128, N,
                (long long)128 * N, 0, 1, N, /*transpose*/1, B, stream);
    launch_gemm(w4, h1024T, g4, b4, fc4, 512, 1024, B, 0, 0, 1, 0, 0, 1, stream);
    launch_gemm(w5, fc4, g5, b5, fc5, 256, 512, B, 0, 0, 1, 0, 0, 1, stream);
    launch_gemm(wt, fc5, nullptr, bt, tbuf, 9, 256, B, 0, 0, 0, 0, 0, 1, stream);
    transform_kernel<<<gcd1((long long)B * 3 * N), B256, 0, stream>>>(x_in, tbuf, xT, B, N);

    // ===================== ec0 on node0 = xT (C=3) =====================
    knn_kernel<<<B * N, B256, 0, stream>>>(xT, xT, 3, N, N, KK, idxb);
    {
        long long tot = (long long)B * 6 * N * KK;
        graph_feature_kernel<<<gcd1(tot), B256, 0, stream>>>(xT, idxb, sB, B, 3, N, KK);
    }
    launch_gemm(e0w0, sB, e0g0, e0b0, sA, 64, 6, N * KK,
                (long long)6 * N * KK, (long long)64 * N * KK, 1, 0, 0, B, stream);
    FILL(x0, (long long)B * 64 * N);
    launch_gemm(e0w1, sA, e0g1, e0b1, x0, 64, 64, N * KK,
                (long long)64 * N * KK, (long long)64 * N, 1, KK, 0, B, stream);
    FILL(xt0, (long long)B * 1024);
    launch_gemm(p0w, x0, p0g, p0b, xt0, 1024, 64, N, (long long)64 * N, 1024, 1, N, 0, B, stream);

    // ===================== randpool 1 + ec1 (C=64) =====================
    gather_kernel<<<gcd1((long long)B * 3 * N1), B256, 0, stream>>>(xT, sidx1, node1, B, 3, N, N1);
    knn_kernel<<<B * N1, B256, 0, stream>>>(node1, xT, 3, N1, N, KK, idxb);
    gather_max_kernel<<<gcd1((long long)B * 64 * N1), B256, 0, stream>>>(x0, idxb, f1, B, 64, N, N1, KK);
    knn_kernel<<<B * N1, B256, 0, stream>>>(f1, f1, 64, N1, N1, KK, idxb);
    {
        long long tot = (long long)B * 128 * N1 * KK;  // 84MB @ sA
        graph_feature_kernel<<<gcd1(tot), B256, 0, stream>>>(f1, idxb, sA, B, 64, N1, KK);
    }
    launch_gemm(e1w0, sA, e1g0, e1b0, sAat(88), 64, 128, N1 * KK,
                (long long)128 * N1 * KK, (long long)64 * N1 * KK, 1, 0, 0, B, stream);
    FILL(x1, (long long)B * 64 * N1);
    launch_gemm(e1w1, sAat(88), e1g1, e1b1, x1, 64, 64, N1 * KK,
                (long long)64 * N1 * KK, (long long)64 * N1, 1, KK, 0, B, stream);
    FILL(xt1, (long long)B * 1024);
    launch_gemm(p1w, x1, p1g, p1b, xt1, 1024, 64, N1, (long long)64 * N1, 1024, 1, N1, 0, B, stream);
    add_relu_kernel<<<gcd1((long long)B * 64 * N1), B256, 0, stream>>>(x1, f1, (long long)B * 64 * N1);

    // ===================== randpool 2 + ec2 =====================
    gather_kernel<<<gcd1((long long)B * 3 * N2), B256, 0, stream>>>(node1, sidx2, node2, B, 3, N1, N2);
    knn_kernel<<<B * N2, B256, 0, stream>>>(node2, node1, 3, N2, N1, KK, idxb);
    gather_max_kernel<<<gcd1((long long)B * 64 * N2), B256, 0, stream>>>(x1, idxb, f2, B, 64, N1, N2, KK);
    knn_kernel<<<B * N2, B256, 0, stream>>>(f2, f2, 64, N2, N2, KK, idxb);
    {
        long long tot = (long long)B * 128 * N2 * KK;
        graph_feature_kernel<<<gcd1(tot), B256, 0, stream>>>(f2, idxb, sA, B, 64, N2, KK);
    }
    launch_gemm(e2w0, sA, e2g0, e2b0, sAat(24), 64, 128, N2 * KK,
                (long long)128 * N2 * KK, (long long)64 * N2 * KK, 1, 0, 0, B, stream);
    FILL(x2, (long long)B * 64 * N2);
    launch_gemm(e2w1, sAat(24), e2g1, e2b1, x2, 64, 64, N2 * KK,
                (long long)64 * N2 * KK, (long long)64 * N2, 1, KK, 0, B, stream);
    FILL(xt2, (long long)B * 1024);
    launch_gemm(p2w, x2, p2g, p2b, xt2, 1024, 64, N2, (long long)64 * N2, 1024, 1, N2, 0, B, stream);
    add_relu_kernel<<<gcd1((long long)B * 64 * N2), B256, 0, stream>>>(x2, f2, (long long)B * 64 * N2);

    // ===================== randpool 3 + ec3 (k = K/2 = 20) =====================
    const int K3 = KK / 2;
    gather_kernel<<<gcd1((long long)B * 3 * N3), B256, 0, stream>>>(node2, sidx3, node3, B, 3, N2, N3);
    knn_kernel<<<B * N3, B256, 0, stream>>>(node3, node2, 3, N3, N2, KK, idxb);
    gather_max_kernel<<<gcd1((long long)B * 64 * N3), B256, 0, stream>>>(x2, idxb, f3, B, 64, N2, N3, KK);
    knn_kernel<<<B * N3, B256, 0, stream>>>(f3, f3, 64, N3, N3, K3, idxb);
    {
        long long tot = (long long)B * 128 * N3 * K3;
        graph_feature_kernel<<<gcd1(tot), B256, 0, stream>>>(f3, idxb, sA, B, 64, N3, K3);
    }
    launch_gemm(e3w0, sA, e3g0, e3b0, sAat(4), 64, 128, N3 * K3,
                (long long)128 * N3 * K3, (long long)64 * N3 * K3, 1, 0, 0, B, stream);
    FILL(x3, (long long)B * 64 * N3);
    launch_gemm(e3w1, sAat(4), e3g1, e3b1, x3, 64, 64, N3 * K3,
                (long long)64 * N3 * K3, (long long)64 * N3, 1, K3, 0, B, stream);
    FILL(xt3, (long long)B * 1024);
    launch_gemm(p3w, x3, p3g, p3b, xt3, 1024, 64, N3, (long long)64 * N3, 1024, 1, N3, 0, B, stream);
    add_relu_kernel<<<gcd1((long long)B * 64 * N3), B256, 0, stream>>>(x3, f3, (long long)B * 64 * N3);

    // ===================== global feature + label =====================
    max4_kernel<<<gcd1((long long)B * 1024), B256, 0, stream>>>(xt0, xt1, xt2, xt3, gmax, (long long)B * 1024);
    launch_gemm(lw, l_in, lg, lb, lf, 64, 16, 1, 16, 64, 1, 0, 0, B, stream);

    // ===================== decode: pn4 =====================
    float* cat4 = sB;                        // (B,1152,32)
    catg_kernel<<<gcd1((long long)B * 1152 * N3), B256, 0, stream>>>(gmax, lf, x3, cat4, B, N3);
    launch_gemm(p4w, cat4, p4g, p4b, h4, 256, 1152, N3,
                (long long)1152 * N3, (long long)256 * N3, 1, 0, 0, B, stream);

    // unpool node3 -> node2, concat x2, pn5
    knn_kernel<<<B * N2, B256, 0, stream>>>(node2, node3, 3, N2, N3, 1, idxb);
    float* up5 = sB;                         // (B,256,128)
    gather_kernel<<<gcd1((long long)B * 256 * N2), B256, 0, stream>>>(h4, idxb, up5, B, 256, N3, N2);
    float* cat5 = sBat(2);                   // (B,320,128)
    cat2_kernel<<<gcd1((long long)B * 320 * N2), B256, 0, stream>>>(up5, x2, cat5, B, 256, 64, N2);
    launch_gemm(p5w, cat5, p5g, p5b, h5, 256, 320, N2,
                (long long)320 * N2, (long long)256 * N2, 1, 0, 0, B, stream);

    // unpool node2 -> node1, concat x1, pn6
    knn_kernel<<<B * N1, B256, 0, stream>>>(node1, node2, 3, N1, N2, 1, idxb);
    float* up6 = sB;                         // (B,256,512)
    gather_kernel<<<gcd1((long long)B * 256 * N1), B256, 0, stream>>>(h5, idxb, up6, B, 256, N2, N1);
    float* cat6 = sBat(8);                   // (B,320,512)
    cat2_kernel<<<gcd1((long long)B * 320 * N1), B256, 0, stream>>>(up6, x1, cat6, B, 256, 64, N1);
    launch_gemm(p6w, cat6, p6g, p6b, h6, 256, 320, N1,
                (long long)320 * N1, (long long)256 * N1, 1, 0, 0, B, stream);

    // unpool node1 -> node0, concat x0, pn7
    knn_kernel<<<B * N, B256, 0, stream>>>(xT, node1, 3, N, N1, 1, idxb);
    float* up7 = sB;                         // (B,256,2048) ~16MB
    gather_kernel<<<gcd1((long long)B * 256 * N), B256, 0, stream>>>(h6, idxb, up7, B, 256, N1, N);
    float* cat7 = sBat(17);                  // (B,320,2048) ~21MB
    cat2_kernel<<<gcd1((long long)B * 320 * N), B256, 0, stream>>>(up7, x0, cat7, B, 256, 64, N);
    launch_gemm(p7w, cat7, p7g, p7b, h7, 128, 320, N,
                (long long)320 * N, (long long)128 * N, 1, 0, 0, B, stream);

    // final linear: w8 (50 x 128), no bn, no act -> d_out (B,50,2048)
    launch_gemm(w8, h7, nullptr, nullptr, (float*)d_out, 50, 128, N,
                (long long)128 * N, (long long)50 * N, 0, 0, 0, B, stream);
}